// EdgeEncoding_15779709846380
// MI455X (gfx1250) — compile-verified
//
#include <hip/hip_runtime.h>

// Problem constants (from reference): B=16, E=2048, P=16384, L=5, D=64
#define BATCH 16
#define NEDGE 2048
#define NPAIR 16384
#define PLEN  5
#define EDIM  64
#define EPSV  1e-9f

typedef float v2f __attribute__((ext_vector_type(2)));
typedef float v8f __attribute__((ext_vector_type(8)));

static constexpr int N_TILES = (BATCH * NPAIR) / 16;   // 16384 tiles of 16 paths
static constexpr int CHUNKS  = (PLEN * EDIM) / 4;      // 80 K-chunks of 4

// Zero a D-length row in workspace; invalid path slots gather from here.
__global__ void EdgeEnc_zero_row(float* z) { z[threadIdx.x] = 0.0f; }

__global__ __launch_bounds__(256) void EdgeEnc_wmma_kernel(
    const float* __restrict__ emb,    // (B, E, D) f32
    const int*   __restrict__ paths,  // (B, P, L) int32, -1 = pad
    const float* __restrict__ ev,     // (L, D) f32
    const float* __restrict__ zrow,   // 64 zero floats in d_ws
    float*       __restrict__ out)    // (B, P) f32
{
    // Pre-masked A-matrix image for V_WMMA_F32_16X16X4_F32:
    // A[r][k] = (r==0) ? ev_flat[4c+k] : 0. Per ISA layout (16x4 f32 A):
    //   VGPR0: lanes0-15 K=0, lanes16-31 K=2 ; VGPR1: K=1 / K=3
    // -> lane 0 holds {ev[4c],ev[4c+1]}, lane 16 holds {ev[4c+2],ev[4c+3]},
    //    all other lanes zero. One ds_load_b64 per chunk in the hot loop.
    __shared__ v2f ldsA[CHUNKS * 32];   // 20 KB

    const int lane = threadIdx.x & 31;
    const int warp = threadIdx.x >> 5;
    const int m    = lane & 15;         // path-in-tile (also output column N)
    const int h    = lane >> 4;         // half: K offset 0 or 2

    for (int t = threadIdx.x; t < CHUNKS * 32; t += blockDim.x) {
        const int c  = t >> 5;
        const int ln = t & 31;
        v2f a; a.x = 0.0f; a.y = 0.0f;
        if ((ln & 15) == 0) {
            const int k = 4 * c + 2 * (ln >> 4);
            a.x = ev[k];
            a.y = ev[k + 1];
        }
        ldsA[t] = a;
    }
    __syncthreads();

    const int waves_total = gridDim.x * (blockDim.x >> 5);
    const int wave_id     = blockIdx.x * (blockDim.x >> 5) + warp;

    for (int tile = wave_id; tile < N_TILES; tile += waves_total) {
        // Global flattened path id of this lane's column (tiles are 16
        // consecutive flat (b*P+p) ids; P % 16 == 0 so tiles never cross b).
        const long pbase = (long)tile * 16 + m;
        const int  b     = tile / (NPAIR / 16);
        const float* embb = emb + (size_t)b * NEDGE * EDIM;

        // Row pointers per level; padding redirects to the zero row.
        const float* rp[PLEN];
        int cnt = 0;
        #pragma unroll
        for (int l = 0; l < PLEN; ++l) {
            const int e      = paths[pbase * PLEN + l];
            const bool valid = (e >= 0);
            cnt += valid ? 1 : 0;
            const float* base = valid ? (embb + (size_t)e * EDIM) : zrow;
            rp[l] = base + 2 * h;       // this half's K offset within a chunk
        }

        // D = sum over 80 chunks of A(ev) x B(gathered), f32 accumulate.
        v8f acc = {};
        #pragma unroll
        for (int l = 0; l < PLEN; ++l) {
            const float* r = rp[l];
            #pragma unroll
            for (int cd = 0; cd < EDIM / 4; ++cd) {
                // B 4x16: lane(h,m) supplies G[m][4c+2h], G[m][4c+2h+1]
                const v2f bv = *(const v2f*)(r + 4 * cd);
                const v2f av = ldsA[(l * (EDIM / 4) + cd) * 32 + lane];
                acc = __builtin_amdgcn_wmma_f32_16x16x4_f32(
                    /*neg_a=*/false, av, /*neg_b=*/false, bv,
                    /*c_mod=*/(short)0, acc, /*reuse_a=*/false, /*reuse_b=*/false);
            }
        }

        // Row 0 of D = results: VGPR0 (acc[0]) on lanes 0-15, N = lane.
        if (lane < 16) {
            const float len = (float)cnt + EPSV;
            out[(long)tile * 16 + lane] = acc[0] / len;
        }
    }
}

extern "C" void kernel_launch(void* const* d_in, const int* in_sizes, int n_in,
                              void* d_out, int out_size, void* d_ws, size_t ws_size,
                              hipStream_t stream) {
    const float* emb   = (const float*)d_in[0];  // (B,E,D) f32
    const int*   paths = (const int*)  d_in[1];  // (B,P,L) int32
    const float* ev    = (const float*)d_in[2];  // (L,D) f32
    float*       out   = (float*)d_out;          // (B,P) f32
    float*       zrow  = (float*)d_ws;           // 256 B zero row

    EdgeEnc_zero_row<<<1, EDIM, 0, stream>>>(zrow);
    EdgeEnc_wmma_kernel<<<1024, 256, 0, stream>>>(emb, paths, ev, zrow, out);
}